// SpikeFP32Softmax_11450382811511
// MI455X (gfx1250) — compile-verified
//
#include <hip/hip_runtime.h>

// SpikeFP32Softmax for MI455X (gfx1250, wave32).
//
// Traffic: 256 MB in + 256 MB out, single pass -> ~22 us at 23.3 TB/s.
// Decode = wave32 ballot + bitreverse (1 coalesced 128B load / value / wave).
// Encode = per-lane bit extract of a wave-uniform u (1 coalesced 128B store).
// Row sum: per-wave uniform accumulation (free) + one V_WMMA_F32_16X16X4_F32
// to reduce the 8 wave partials (A rows 0-1 hold the partials, B = ones).

typedef float v2f __attribute__((ext_vector_type(2)));
typedef float v8f __attribute__((ext_vector_type(8)));

#define ROWS 1024
#define NVAL 2048
#define BITS 32
#define TPB 256
#define WAVES (TPB / 32)            // 8 waves per block
#define VPW (NVAL / WAVES)          // 256 values per wave

__global__ __launch_bounds__(TPB)
void spike_fp32_softmax_kernel(const float* __restrict__ x,
                               float* __restrict__ out) {
    __shared__ float lds_e[NVAL];   // exp(v) per value, 8 KB
    __shared__ float lds_p[WAVES];  // per-wave partial sums

    const int row  = blockIdx.x;
    const int lane = threadIdx.x & 31;
    const int wave = threadIdx.x >> 5;

    const float* __restrict__ xrow = x   + (size_t)row * NVAL * BITS;
    float*       __restrict__ orow = out + (size_t)row * NVAL * BITS;

    const int vbase = wave * VPW;

    // ---- Phase 1: decode bits -> fp32, exp, wave-uniform partial sum ----
    float psum = 0.0f;
    #pragma unroll 4
    for (int i = 0; i < VPW; ++i) {
        const int v = vbase + i;
        // lane j holds bit j (MSB-first) of value v: coalesced 128B load
        float b = xrow[(size_t)v * BITS + lane];
        unsigned mask = __builtin_amdgcn_ballot_w32(b != 0.0f); // bit L = lane L
        unsigned u = __builtin_bitreverse32(mask);              // MSB-first -> u
        float val = __uint_as_float(u);
        float e = __expf(val);                                  // v_exp_f32 (TRANS)
        psum += e;                                              // wave-uniform
        if (lane == 0) lds_e[v] = e;
    }
    if (lane == 0) lds_p[wave] = psum;
    __syncthreads();

    // ---- Phase 2: reduce 8 partials with V_WMMA_F32_16X16X4_F32 ----
    // A (16x4 f32) layout: lanes 0-15 -> rows 0-15, VGPR0=K0,VGPR1=K1;
    //                      lanes 16-31 -> rows 0-15, VGPR0=K2,VGPR1=K3.
    // Put p0..p3 in A[0][0..3], p4..p7 in A[1][0..3]; B = ones ->
    // D[0][n] = p0+p1+p2+p3, D[1][n] = p4+p5+p6+p7 for every column n.
    {
        const int r  = lane & 15;        // A-matrix row
        const int kp = lane >> 4;        // 0 -> K=0,1 ; 1 -> K=2,3
        const int bi = (r * 4 + kp * 2) & 7;     // clamp (rows >=2 unused)
        const bool live = (r < 2);
        float a0 = live ? lds_p[bi]     : 0.0f;  // branchless -> EXEC all-ones
        float a1 = live ? lds_p[bi + 1] : 0.0f;
        v2f A = {a0, a1};
        v2f B = {1.0f, 1.0f};
        v8f C = {};
        v8f D = __builtin_amdgcn_wmma_f32_16x16x4_f32(
            /*neg_a=*/false, A, /*neg_b=*/false, B,
            /*c_mod=*/(short)0, C, /*reuse_a=*/false, /*reuse_b=*/false);
        // lanes 0-15: D[0]=M0 row, D[1]=M1 row -> total sum
        float s = D[0] + D[1];
        // broadcast lane 0's value to the whole wave
        s = __uint_as_float((unsigned)__builtin_amdgcn_readfirstlane(
                (int)__float_as_uint(s)));
        lds_p[wave] = s;  // stash uniform total (any lane; same value)
    }
    const float inv_s = 1.0f / lds_p[wave];

    // ---- Phase 3: normalize + encode bits, coalesced 128B stores ----
    const unsigned sh = 31u - (unsigned)lane;
    #pragma unroll 4
    for (int i = 0; i < VPW; ++i) {
        const int v = vbase + i;
        float rr = lds_e[v] * inv_s;          // LDS broadcast read
        unsigned u = __float_as_uint(rr);     // wave-uniform bits
        orow[(size_t)v * BITS + lane] = (float)((u >> sh) & 1u);
    }
}

extern "C" void kernel_launch(void* const* d_in, const int* in_sizes, int n_in,
                              void* d_out, int out_size, void* d_ws, size_t ws_size,
                              hipStream_t stream) {
    const float* x = (const float*)d_in[0];
    float* out = (float*)d_out;
    (void)in_sizes; (void)n_in; (void)out_size; (void)d_ws; (void)ws_size;
    spike_fp32_softmax_kernel<<<ROWS, TPB, 0, stream>>>(x, out);
}